// Net_GIN_11227044511900
// MI455X (gfx1250) — compile-verified
//
#include <hip/hip_runtime.h>
#include <math.h>

typedef __attribute__((ext_vector_type(2))) float v2f;
typedef __attribute__((ext_vector_type(8))) float v8f;

#define N_NODES   100000
#define N_EDGES   1000000
#define NHID      64
#define N_CLASSES 10
#define N_GRAPHS  512
#define BN_EPS    1e-5f

// ---------------------------------------------------------------- copy h -> agg
__global__ void copy_f4(const float4* __restrict__ in, float4* __restrict__ out, int n) {
    int i = blockIdx.x * blockDim.x + threadIdx.x;
    if (i < n) out[i] = in[i];
}

// ------------------------------------------- edge scatter: agg[dst] += h[src]
// 16 lanes per edge, each lane handles 4 consecutive floats (float4 gather).
// Working set is L2-resident (25.6MB per node tensor, 192MB L2).
__global__ void edge_scatter(const float* __restrict__ h, const int* __restrict__ src,
                             const int* __restrict__ dst, float* __restrict__ agg) {
    int t = blockIdx.x * blockDim.x + threadIdx.x;
    int e = t >> 4, f = t & 15;
    if (e >= N_EDGES) return;
    int s = src[e], d = dst[e];
    float4 v = ((const float4*)(h + (size_t)s * NHID))[f];
    float* p = agg + (size_t)d * NHID + f * 4;
    atomicAdd(p + 0, v.x);
    atomicAdd(p + 1, v.y);
    atomicAdd(p + 2, v.z);
    atomicAdd(p + 3, v.w);
}

// -------------------------------------------------- pooled[batch[i]] += h[i]
__global__ void pool_scatter(const float* __restrict__ h, const int* __restrict__ batch,
                             float* __restrict__ pooled) {
    int t = blockIdx.x * blockDim.x + threadIdx.x;
    int node = t >> 4, f = t & 15;
    if (node >= N_NODES) return;
    int g = batch[node];
    float4 v = ((const float4*)(h + (size_t)node * NHID))[f];
    float* p = pooled + (size_t)g * NHID + f * 4;
    atomicAdd(p + 0, v.x);
    atomicAdd(p + 1, v.y);
    atomicAdd(p + 2, v.z);
    atomicAdd(p + 3, v.w);
}

// -------------------------------------------------------- WMMA f32 GEMM 64x64
// out[nrows x 64] = epilogue(A[nrows x 64] @ W[64 x 64])
// mode 0: BN(x + bias) then ReLU  (folded into one FMA per element)
// mode 1: bias + ReLU
// 8 waves per block; wave w -> 16x16 tile (row half = w&1, col tile = w>>1)
// of a 32-row x 64-col output block. K=64 done as 16x V_WMMA_F32_16X16X4_F32.
// Layouts per CDNA5 ISA 7.12.2:
//   A 16x4 f32 : lane L holds A[L&15][kbase], A[L&15][kbase+1], kbase=(L>>4)*2
//   B 4x16 f32 : lane L holds B[kbase][L&15], B[kbase+1][L&15]
//   C/D 16x16  : VGPR j, lane L -> C[j + (L>>4)*8][L&15]
__global__ void __launch_bounds__(256)
gemm64_wmma(const float* __restrict__ A, const float* __restrict__ W,
            const float* __restrict__ bias,
            const float* __restrict__ gamma, const float* __restrict__ beta,
            const float* __restrict__ mean, const float* __restrict__ var,
            float* __restrict__ out, int nrows, int mode) {
    const int lane = threadIdx.x & 31;
    const int wave = threadIdx.x >> 5;
    const int half = lane >> 4;
    const int l15  = lane & 15;
    const int tr   = wave & 1;
    const int n0   = (wave >> 1) * 16;
    const int n    = n0 + l15;

    // Preload all B fragments for this column tile (16 K-steps x float2 = 32 VGPRs)
    v2f bfrag[16];
#pragma unroll
    for (int j = 0; j < 16; ++j) {
        int k = j * 4 + half * 2;
        v2f b;
        b.x = W[k * NHID + n];
        b.y = W[(k + 1) * NHID + n];
        bfrag[j] = b;
    }

    // Per-column epilogue coefficients: val = max(acc*s + t, 0)
    float s, t;
    if (mode == 0) {
        float inv = rsqrtf(var[n] + BN_EPS);
        s = gamma[n] * inv;
        t = (bias[n] - mean[n]) * s + beta[n];
    } else {
        s = 1.0f;
        t = bias[n];
    }

    const int nblk = nrows >> 5;   // nrows is a multiple of 32 (100000, 512)
    for (int rb = blockIdx.x; rb < nblk; rb += gridDim.x) {
        int row0 = rb * 32 + tr * 16;
        v8f acc = {};
#pragma unroll
        for (int j = 0; j < 16; ++j) {
            int k = j * 4 + half * 2;
            const float* ap = A + (size_t)(row0 + l15) * NHID + k;
            v2f a;
            a.x = ap[0];
            a.y = ap[1];
            acc = __builtin_amdgcn_wmma_f32_16x16x4_f32(
                /*neg_a=*/false, a, /*neg_b=*/false, bfrag[j],
                /*c_mod=*/(short)0, acc, /*reuse_a=*/false, /*reuse_b=*/false);
        }
#pragma unroll
        for (int jj = 0; jj < 8; ++jj) {
            int row = row0 + jj + half * 8;
            float v = acc[jj] * s + t;
            out[(size_t)row * NHID + n] = v > 0.0f ? v : 0.0f;
        }
    }
}

// ------------------------------------------- lin2 (64x10) + log_softmax head
__global__ void head_kernel(const float* __restrict__ h, const float* __restrict__ w,
                            const float* __restrict__ b, float* __restrict__ out) {
    int r = blockIdx.x * blockDim.x + threadIdx.x;
    if (r >= N_GRAPHS) return;
    float logit[N_CLASSES];
#pragma unroll
    for (int c = 0; c < N_CLASSES; ++c) logit[c] = b[c];
    for (int k = 0; k < NHID; ++k) {
        float hv = h[(size_t)r * NHID + k];
#pragma unroll
        for (int c = 0; c < N_CLASSES; ++c) logit[c] += hv * w[k * N_CLASSES + c];
    }
    float mx = logit[0];
#pragma unroll
    for (int c = 1; c < N_CLASSES; ++c) mx = fmaxf(mx, logit[c]);
    float sum = 0.0f;
#pragma unroll
    for (int c = 0; c < N_CLASSES; ++c) sum += __expf(logit[c] - mx);
    float lse = mx + __logf(sum);
#pragma unroll
    for (int c = 0; c < N_CLASSES; ++c) out[(size_t)r * N_CLASSES + c] = logit[c] - lse;
}

extern "C" void kernel_launch(void* const* d_in, const int* in_sizes, int n_in,
                              void* d_out, int out_size, void* d_ws, size_t ws_size,
                              hipStream_t stream) {
    const float* x     = (const float*)d_in[0];
    const int*   ei    = (const int*)d_in[1];
    const int*   src   = ei;
    const int*   dst   = ei + N_EDGES;
    const int*   batch = (const int*)d_in[2];
    const float* cw1   = (const float*)d_in[3];
    const float* cb1   = (const float*)d_in[4];
    const float* bng   = (const float*)d_in[5];
    const float* bnb   = (const float*)d_in[6];
    const float* bnm   = (const float*)d_in[7];
    const float* bnv   = (const float*)d_in[8];
    const float* cw2   = (const float*)d_in[9];
    const float* cb2   = (const float*)d_in[10];
    const float* l1w   = (const float*)d_in[11];
    const float* l1b   = (const float*)d_in[12];
    const float* l2w   = (const float*)d_in[13];
    const float* l2b   = (const float*)d_in[14];

    char* ws = (char*)d_ws;
    const size_t NB = (size_t)N_NODES * NHID * sizeof(float);   // 25.6 MB
    float* agg    = (float*)(ws);
    float* tmp    = (float*)(ws + NB);
    float* hbuf   = (float*)(ws + 2 * NB);
    float* pooled = (float*)(ws + 3 * NB);
    float* l1out  = (float*)(ws + 3 * NB + 64 * 1024);

    const dim3 blk(256);
    const int nf4      = N_NODES * (NHID / 4);                 // 1.6M float4
    const int copyGrid = (nf4 + 255) / 256;
    const int edgeGrid = (N_EDGES * 16 + 255) / 256;
    const int poolGrid = (N_NODES * 16 + 255) / 256;
    const int gemmGrid = N_NODES / 32;                         // 3125 blocks

    const float* hcur = x;
    for (int l = 0; l < 3; ++l) {
        // agg = h  (self term, eps=0), then agg[dst] += h[src]
        copy_f4<<<copyGrid, blk, 0, stream>>>((const float4*)hcur, (float4*)agg, nf4);
        edge_scatter<<<edgeGrid, blk, 0, stream>>>(hcur, src, dst, agg);
        // tmp = relu(BN(agg @ w1 + b1))
        gemm64_wmma<<<gemmGrid, blk, 0, stream>>>(
            agg, cw1 + (size_t)l * NHID * NHID, cb1 + l * NHID,
            bng + l * NHID, bnb + l * NHID, bnm + l * NHID, bnv + l * NHID,
            tmp, N_NODES, 0);
        // hbuf = relu(tmp @ w2 + b2)
        gemm64_wmma<<<gemmGrid, blk, 0, stream>>>(
            tmp, cw2 + (size_t)l * NHID * NHID, cb2 + l * NHID,
            nullptr, nullptr, nullptr, nullptr,
            hbuf, N_NODES, 1);
        hcur = hbuf;
    }

    // global_add_pool
    hipMemsetAsync(pooled, 0, (size_t)N_GRAPHS * NHID * sizeof(float), stream);
    pool_scatter<<<poolGrid, blk, 0, stream>>>(hcur, batch, pooled);

    // lin1 + ReLU (512x64 @ 64x64) — same WMMA kernel
    gemm64_wmma<<<N_GRAPHS / 32, blk, 0, stream>>>(
        pooled, l1w, l1b, nullptr, nullptr, nullptr, nullptr,
        l1out, N_GRAPHS, 1);

    // lin2 + log_softmax
    head_kernel<<<(N_GRAPHS + 255) / 256, blk, 0, stream>>>(l1out, l2w, l2b, (float*)d_out);
}